// SphericalFilter_21835613733174
// MI455X (gfx1250) — compile-verified
//
#include <hip/hip_runtime.h>
#include <math.h>

typedef __attribute__((ext_vector_type(2))) float v2f;
typedef __attribute__((ext_vector_type(8))) float v8f;

#define SF_MAXF    64
#define SF_NFEAT   129           // 2*MAXF + 1
#define SF_FPAD    132           // padded to multiple of 4 (K padding, zeros)
#define SF_HIDDEN  256
#define SF_PTS_BLK 64            // points per workgroup (4 M-tiles of 16)

// out[row, h] = sum_f Y[row, f] * W[h, f] + b[h]
// Y built on the fly:   Y[64]    = sqrt(3/4pi)*cos(phi)
//                       Y[64+k]  = D_k * cos(phi) * sin(phi)^k * cos(k*theta)
//                       Y[64-k]  = D_k * cos(phi) * sin(phi)^k * sin(k*theta)
// with D_1 = sqrt(15/4pi), D_k = D_{k-1} * sqrt((2k+3)/(2k)).
__global__ __launch_bounds__(256) void sph_filter_wmma_kernel(
    const float* __restrict__ x,     // [B*N, 2] (theta, phi)
    const float* __restrict__ W,     // [256, 129] row-major
    const float* __restrict__ bias,  // [256]
    float* __restrict__ out)         // [B*N, 256]
{
  __shared__ float Ys[SF_PTS_BLK * SF_FPAD];   // 33792 B
  const int tid = threadIdx.x;

  // ---------------- Phase 1: features, one point per thread (threads 0..63)
  if (tid < SF_PTS_BLK) {
    const long g = (long)blockIdx.x * SF_PTS_BLK + tid;
    const float theta = x[2 * g + 0];
    const float phi   = x[2 * g + 1];
    float ct, st, cphi, sphi;
    sincosf(theta, &st, &ct);
    sincosf(phi, &sphi, &cphi);

    float* Yr = &Ys[tid * SF_FPAD];
    Yr[SF_MAXF] = 0.48860251190291992f * cphi;   // sqrt(3/(4pi)) * cos(phi)
    Yr[SF_NFEAT + 0] = 0.0f;                     // K padding (129..131)
    Yr[SF_NFEAT + 1] = 0.0f;
    Yr[SF_NFEAT + 2] = 0.0f;

    float ck = ct, sk = st;              // cos(k*theta), sin(k*theta)
    float sp = sphi;                     // sin(phi)^k
    float Dk = 1.0925484305920792f;      // sqrt(15/(4pi))
    #pragma unroll 4
    for (int k = 1; k <= SF_MAXF; ++k) {
      const float base = Dk * cphi * sp;
      Yr[SF_MAXF + k] = base * ck;
      Yr[SF_MAXF - k] = base * sk;
      const float cn = ck * ct - sk * st;          // angle-addition recurrence
      sk = sk * ct + ck * st;
      ck = cn;
      sp *= sphi;
      const float fk = (float)k;
      Dk *= sqrtf((2.0f * fk + 5.0f) / (2.0f * fk + 2.0f));
    }
  }
  __syncthreads();

  // ---------------- Phase 2: WMMA GEMM (full EXEC from here on)
  const int wave = tid >> 5;             // 0..7, owns hidden cols [32w, 32w+32)
  const int lane = tid & 31;
  const int n    = lane & 15;            // N index inside 16x16 tile
  const int koff = (lane >> 4) << 1;     // K sub-offset per A/B f32 layout: 0 or 2
  const int h0   = wave * 32;

  const v8f vzero = {0.f, 0.f, 0.f, 0.f, 0.f, 0.f, 0.f, 0.f};
  v8f acc[4][2];
  #pragma unroll
  for (int mt = 0; mt < 4; ++mt)
    #pragma unroll
    for (int ht = 0; ht < 2; ++ht)
      acc[mt][ht] = vzero;

  // chunks 0..31: all K in-bounds for W
  for (int c = 0; c < 32; ++c) {
    const int f0 = 4 * c + koff;
    v2f bfrag[2];
    #pragma unroll
    for (int ht = 0; ht < 2; ++ht) {
      const float* wp = &W[(h0 + 16 * ht + n) * SF_NFEAT + f0];
      bfrag[ht].x = wp[0];
      bfrag[ht].y = wp[1];
    }
    #pragma unroll
    for (int mt = 0; mt < 4; ++mt) {
      const v2f afrag = *(const v2f*)&Ys[(16 * mt + n) * SF_FPAD + f0];
      #pragma unroll
      for (int ht = 0; ht < 2; ++ht) {
        acc[mt][ht] = __builtin_amdgcn_wmma_f32_16x16x4_f32(
            false, afrag, false, bfrag[ht], (short)0, acc[mt][ht], false, false);
      }
    }
  }
  // last chunk (K = 128..131): only f=128 valid in W; A rows are zero-padded
  {
    const int f0 = 128 + koff;
    v2f bfrag[2];
    #pragma unroll
    for (int ht = 0; ht < 2; ++ht) {
      const float w128 = W[(h0 + 16 * ht + n) * SF_NFEAT + 128];  // in-bounds all lanes
      bfrag[ht].x = (koff == 0) ? w128 : 0.0f;                    // lane select, no OOB
      bfrag[ht].y = 0.0f;
    }
    #pragma unroll
    for (int mt = 0; mt < 4; ++mt) {
      const v2f afrag = *(const v2f*)&Ys[(16 * mt + n) * SF_FPAD + f0];
      #pragma unroll
      for (int ht = 0; ht < 2; ++ht) {
        acc[mt][ht] = __builtin_amdgcn_wmma_f32_16x16x4_f32(
            false, afrag, false, bfrag[ht], (short)0, acc[mt][ht], false, false);
      }
    }
  }

  // ---------------- Epilogue: bias + store per 16x16 f32 C/D layout
  const long row0  = (long)blockIdx.x * SF_PTS_BLK;
  const int  rhalf = (lane >> 4) * 8;    // lanes 16..31 hold M = r + 8
  #pragma unroll
  for (int ht = 0; ht < 2; ++ht) {
    const int h = h0 + 16 * ht + n;
    const float bv = bias[h];
    #pragma unroll
    for (int mt = 0; mt < 4; ++mt) {
      #pragma unroll
      for (int r = 0; r < 8; ++r) {
        const long row = row0 + 16 * mt + rhalf + r;
        out[row * SF_HIDDEN + h] = acc[mt][ht][r] + bv;
      }
    }
  }
}

extern "C" void kernel_launch(void* const* d_in, const int* in_sizes, int n_in,
                              void* d_out, int out_size, void* d_ws, size_t ws_size,
                              hipStream_t stream) {
  (void)n_in; (void)out_size; (void)d_ws; (void)ws_size;
  const float* x    = (const float*)d_in[0];   // [B, N, 2] f32
  const float* W    = (const float*)d_in[1];   // [256, 129] f32
  const float* bias = (const float*)d_in[2];   // [256] f32
  float* out = (float*)d_out;                  // [B, N, 256] f32

  const int npts = in_sizes[0] / 2;            // B*N = 262144
  const int nblocks = npts / SF_PTS_BLK;       // 4096
  sph_filter_wmma_kernel<<<dim3(nblocks), dim3(256), 0, stream>>>(x, W, bias, out);
}